// QuantumRecurrentUnit_38293928411479
// MI455X (gfx1250) — compile-verified
//
#include <hip/hip_runtime.h>
#include <hip/hip_bf16.h>

// 11-qubit circuit: wire q <-> bit (10-q) of the 2048-dim state index.
// Group A = wires 0-3 (bits 10..7), B = wires 4-7 (bits 6..3), C = wires 8-10 (bits 2..0).
// Layer factors as  D_{10,0} . M_C . D_{7,8} . M_B . D_{3,4} . M_A  (right-to-left),
// with the diagonal CZ signs folded into the NEXT matmul's fragment loads.

#define NQ 11
#define QDIM 2048

typedef float v2f __attribute__((ext_vector_type(2)));
typedef float v4f __attribute__((ext_vector_type(4)));
typedef float v8f __attribute__((ext_vector_type(8)));

__device__ __forceinline__ v8f wmma_k4(v2f a, v2f b, v8f c) {
  // D(16x16,f32) += A(16x4,f32) x B(4x16,f32)
  return __builtin_amdgcn_wmma_f32_16x16x4_f32(false, a, false, b, (short)0, c,
                                               false, false);
}

// ---------------------------------------------------------------------------
// Prep kernel: build the three (layer-invariant) group matrices from weights.
//   ws[  0..255] : MA 16x16 row-major   (wires 0-3,  wire j -> group bit 3-j)
//   ws[256..511] : MB 16x16 row-major   (wires 4-7)
//   ws[512..639] : MCpad 16x8 row-major (wires 8-10; rows 8..15 zero)
// ---------------------------------------------------------------------------
__global__ __launch_bounds__(256) void build_gate_mats(
    const float* __restrict__ w, float* __restrict__ ws) {
  __shared__ float M[16][16];
  const int tid = threadIdx.x;
  const int r = tid >> 4, col = tid & 15;

  auto initI = [&]() {
    M[r][col] = (r == col) ? 1.0f : 0.0f;
    __syncthreads();
  };
  // M <- RY(theta on group-bit p) * M   (rows mix in pairs)
  auto ry = [&](float th, int p, int dim) {
    float c = cosf(0.5f * th), s = sinf(0.5f * th);
    bool act = (r < dim) && (col < dim);
    float self = 0.0f, par = 0.0f;
    if (act) { self = M[r][col]; par = M[r ^ (1 << p)][col]; }
    __syncthreads();
    if (act)
      M[r][col] = ((r >> p) & 1) ? (s * par + c * self) : (c * self - s * par);
    __syncthreads();
  };
  // M <- CZ(bits p1,p2) * M  (sign flip rows with both bits set)
  auto cz = [&](int p1, int p2, int dim) {
    bool act = (r < dim) && (col < dim);
    if (act && (((r >> p1) & (r >> p2)) & 1)) M[r][col] = -M[r][col];
    __syncthreads();
  };

  // Group A: RY0 CZ01 RY1 CZ12 RY2 CZ23 RY3 on bits 3..0
  initI();
  ry(w[0], 3, 16); cz(3, 2, 16);
  ry(w[1], 2, 16); cz(2, 1, 16);
  ry(w[2], 1, 16); cz(1, 0, 16);
  ry(w[3], 0, 16);
  ws[tid] = M[r][col];
  __syncthreads();

  // Group B: RY4 CZ45 RY5 CZ56 RY6 CZ67 RY7
  initI();
  ry(w[4], 3, 16); cz(3, 2, 16);
  ry(w[5], 2, 16); cz(2, 1, 16);
  ry(w[6], 1, 16); cz(1, 0, 16);
  ry(w[7], 0, 16);
  ws[256 + tid] = M[r][col];
  __syncthreads();

  // Group C (8x8): RY8 CZ89 RY9 CZ9,10 RY10 on bits 2..0
  initI();
  ry(w[8], 2, 8);  cz(2, 1, 8);
  ry(w[9], 1, 8);  cz(1, 0, 8);
  ry(w[10], 0, 8);
  if (tid < 128) {  // MCpad: 16 rows x 8 cols, bottom half zero
    int rp = tid >> 3, cp = tid & 7;
    ws[512 + tid] = (rp < 8) ? M[rp][cp] : 0.0f;
  }
}

// ---------------------------------------------------------------------------
// Main kernel: one sample per wave, state (2048 f32) resident in LDS.
// 8 waves / block (64KB LDS), 256 blocks -> 2048 samples.
// ---------------------------------------------------------------------------
__global__ __launch_bounds__(256) void qsim_kernel(
    const float* __restrict__ x, const float* __restrict__ ws,
    float* __restrict__ out) {
  __shared__ float sh[8 * QDIM];
  const int lane = threadIdx.x & 31;
  const int wv = threadIdx.x >> 5;
  const int lo = lane & 15;   // fragment row/col within 16
  const int hi = lane >> 4;   // fragment half select
  const int sample = blockIdx.x * 8 + wv;
  float* S = sh + wv * QDIM;

  // ---- encoding: product state amp[b] = prod_q (bit ? sin : cos) ----
  float cs[NQ], sn[NQ];
#pragma unroll
  for (int q = 0; q < NQ; ++q) {
    float h = 0.5f * x[sample * 16 + q];
    cs[q] = cosf(h);
    sn[q] = sinf(h);
  }
  for (int i = 0; i < QDIM / 32; ++i) {
    int b = i * 32 + lane;
    float amp = 1.0f;
#pragma unroll
    for (int q = 0; q < NQ; ++q) amp *= ((b >> (10 - q)) & 1) ? sn[q] : cs[q];
    S[b] = amp;
  }

  // ---- A-operand fragments (ISA 16x4 f32 layout), shared by all 6 layers ----
  v2f mA[4], mB[4], mC[2];
#pragma unroll
  for (int k = 0; k < 4; ++k) {
    int kk = 4 * k + 2 * hi;
    mA[k].x = ws[lo * 16 + kk];       mA[k].y = ws[lo * 16 + kk + 1];
    mB[k].x = ws[256 + lo * 16 + kk]; mB[k].y = ws[256 + lo * 16 + kk + 1];
  }
#pragma unroll
  for (int k = 0; k < 2; ++k) {
    int kk = 4 * k + 2 * hi;
    mC[k].x = ws[512 + lo * 8 + kk];  mC[k].y = ws[512 + lo * 8 + kk + 1];
  }

  for (int layer = 0; layer < 6; ++layer) {
    // ---- Group A: new[a',n] = sum_a MA[a',a] * S[a*128+n], n in 0..127 ----
    // fold previous layer's CZ(10,0): sign if (bit10 & bit0) = (a>>3) & (n&1)
    for (int t = 0; t < 8; ++t) {
      int n = t * 16 + lo;
      v2f bf[4];
#pragma unroll
      for (int k = 0; k < 4; ++k) {
        int a0 = 4 * k + 2 * hi;
        float b0 = S[a0 * 128 + n];
        float b1 = S[(a0 + 1) * 128 + n];
        if (layer > 0) {
          if (((a0 >> 3) & n) & 1) b0 = -b0;
          if ((((a0 + 1) >> 3) & n) & 1) b1 = -b1;
        }
        bf[k].x = b0; bf[k].y = b1;
      }
      v8f d;
#pragma unroll
      for (int v = 0; v < 8; ++v) d[v] = 0.0f;
#pragma unroll
      for (int k = 0; k < 4; ++k) d = wmma_k4(mA[k], bf[k], d);
#pragma unroll
      for (int v = 0; v < 8; ++v) S[(v + 8 * hi) * 128 + n] = d[v];
    }

    // ---- Group B over axis b of idx = a*128 + b*8 + c;  n=(a,c) ----
    // fold CZ(3,4): sign if bit7 & bit6 = (a&1) & (b>>3)
    for (int t = 0; t < 8; ++t) {
      int n = t * 16 + lo;
      int a = n >> 3, cc = n & 7;
      int base = a * 128 + cc;
      v2f bf[4];
#pragma unroll
      for (int k = 0; k < 4; ++k) {
        int b0r = 4 * k + 2 * hi;
        float v0 = S[base + b0r * 8];
        float v1 = S[base + (b0r + 1) * 8];
        if ((a & (b0r >> 3)) & 1) v0 = -v0;
        if ((a & ((b0r + 1) >> 3)) & 1) v1 = -v1;
        bf[k].x = v0; bf[k].y = v1;
      }
      v8f d;
#pragma unroll
      for (int v = 0; v < 8; ++v) d[v] = 0.0f;
#pragma unroll
      for (int k = 0; k < 4; ++k) d = wmma_k4(mB[k], bf[k], d);
#pragma unroll
      for (int v = 0; v < 8; ++v) S[base + (v + 8 * hi) * 8] = d[v];
    }

    // ---- Group C over axis c (K=8, MC padded to 16x8); tile t fixes a=t, b=lo
    // fold CZ(7,8): sign if bit3 & bit2 = (b&1) & (c>>2)
    // contiguous in LDS -> vector ds ops: 2x b64 loads, 2x b128 stores per lane
    for (int t = 0; t < 16; ++t) {
      float* row = S + t * 128 + lo * 8;  // 8 consecutive floats, 32B aligned
      v2f bf[2];
#pragma unroll
      for (int k = 0; k < 2; ++k) {
        int c0 = 4 * k + 2 * hi;
        v2f v = *reinterpret_cast<const v2f*>(row + c0);  // ds_load_b64
        if ((lo & (c0 >> 2)) & 1) v.x = -v.x;
        if ((lo & ((c0 + 1) >> 2)) & 1) v.y = -v.y;
        bf[k] = v;
      }
      v8f d;
#pragma unroll
      for (int v = 0; v < 8; ++v) d[v] = 0.0f;
      d = wmma_k4(mC[0], bf[0], d);
      d = wmma_k4(mC[1], bf[1], d);
      if (hi == 0) {  // rows 8..15 of D are the zero padding; only c'=0..7 valid
        v4f s0, s1;
        s0.x = d[0]; s0.y = d[1]; s0.z = d[2]; s0.w = d[3];
        s1.x = d[4]; s1.y = d[5]; s1.z = d[6]; s1.w = d[7];
        *reinterpret_cast<v4f*>(row) = s0;       // ds_store_b128
        *reinterpret_cast<v4f*>(row + 4) = s1;   // ds_store_b128
      }
    }
  }

  // ---- epilogue: <Z_q> = sum_b S[b]^2 * (1 - 2*bit_{10-q}(b)) ----
  // (trailing CZ(10,0) sign is squared away)
  float acc[NQ];
#pragma unroll
  for (int q = 0; q < NQ; ++q) acc[q] = 0.0f;
  for (int i = 0; i < QDIM / 32; ++i) {
    int b = i * 32 + lane;
    float p = S[b] * S[b];
#pragma unroll
    for (int q = 0; q < NQ; ++q) acc[q] += ((b >> (10 - q)) & 1) ? -p : p;
  }
#pragma unroll
  for (int q = 0; q < NQ; ++q)
    for (int off = 16; off > 0; off >>= 1)
      acc[q] += __shfl_down(acc[q], off, 32);
  if (lane == 0) {
#pragma unroll
    for (int q = 0; q < NQ; ++q) out[sample * NQ + q] = acc[q];
  }
}

extern "C" void kernel_launch(void* const* d_in, const int* in_sizes, int n_in,
                              void* d_out, int out_size, void* d_ws,
                              size_t ws_size, hipStream_t stream) {
  const float* x = (const float*)d_in[0];       // [8,256,16] f32
  const float* w = (const float*)d_in[1];       // [11] f32
  float* out = (float*)d_out;                   // [8,256,11] f32
  float* ws = (float*)d_ws;                     // 640 floats of gate matrices

  build_gate_mats<<<1, 256, 0, stream>>>(w, ws);
  qsim_kernel<<<256, 256, 0, stream>>>(x, ws, out);
}